// AggrHGraphConvLayer_3659312136367
// MI455X (gfx1250) — compile-verified
//
#include <hip/hip_runtime.h>
#include <hip/hip_bf16.h>

typedef __attribute__((ext_vector_type(2))) float v2f;
typedef __attribute__((ext_vector_type(8))) float v8f;

#define FD 128     // feature dim
#define RB 80      // rows per GEMM block (5 x 16-row WMMA tiles); 10000/20000/50000 all % 80 == 0
#define RT 5       // row tiles per wave

// ---------------------------------------------------------------- utilities
__global__ void zero_kernel(float* __restrict__ p, int n) {
    int i = blockIdx.x * blockDim.x + threadIdx.x;
    if (i < n) p[i] = 0.0f;
}

// out_deg[src[e]] += 1 ; in_deg[dst[e]] += 1
__global__ void degree_kernel(const int* __restrict__ src, const int* __restrict__ dst,
                              float* __restrict__ outdeg, float* __restrict__ indeg, int ne) {
    int e = blockIdx.x * blockDim.x + threadIdx.x;
    if (e < ne) {
        atomicAdd(&outdeg[src[e]], 1.0f);
        atomicAdd(&indeg[dst[e]], 1.0f);
    }
}

// ------------------------------------------------------- WMMA fp32 GEMM
// H[r, :] = (X[r, :] @ W) * rsqrt(max(outdeg[r],1))
// Block = 256 threads = 8 waves, covers RB=80 rows. Wave w owns cols [16w,16w+16)
// and RT=5 row tiles, so each pair of B loads feeds 5 WMMAs (B register reuse).
//
// ISA layouts (CDNA5 ISA):
//   A (16x4 f32):  VGPR v, lanes 0-15: M=lane, K=k0+v ; lanes 16-31: M=lane-16, K=k0+v+2
//   B (4x16 f32):  VGPR v, lanes 0-15: K=k0+v, N=lane ; lanes 16-31: K=k0+v+2, N=lane-16
//   C/D (16x16):   VGPR i, lanes 0-15: M=i, N=lane ; lanes 16-31: M=i+8, N=lane-16
__global__ void gemm_outdeg_kernel(const float* __restrict__ X, const float* __restrict__ W,
                                   const float* __restrict__ outdeg, float* __restrict__ H) {
    __shared__ float sx[RB][132];  // 528B row stride: float4-aligned, bank-conflict-free
    const int row0 = blockIdx.x * RB;

    // coalesced float4 load of the 80x128 X tile
    for (int i = threadIdx.x; i < RB * (FD / 4); i += 256) {
        int r = i >> 5, c4 = (i & 31) << 2;
        float4 v = *(const float4*)(X + (long long)(row0 + r) * FD + c4);
        *(float4*)(&sx[r][c4]) = v;
    }
    __syncthreads();

    const int wave = threadIdx.x >> 5;
    const int lane = threadIdx.x & 31;
    const int col0 = wave * 16;
    const int half = lane >> 4;   // 0: lanes 0-15, 1: lanes 16-31
    const int l16  = lane & 15;

    v8f acc[RT];
#pragma unroll
    for (int t = 0; t < RT; ++t) acc[t] = (v8f){};

    for (int k0 = 0; k0 < FD; k0 += 4) {
        const int kb = k0 + 2 * half;
        v2f b;
        b.x = W[(kb + 0) * FD + col0 + l16];
        b.y = W[(kb + 1) * FD + col0 + l16];
#pragma unroll
        for (int t = 0; t < RT; ++t) {
            v2f a;
            a.x = sx[t * 16 + l16][kb + 0];
            a.y = sx[t * 16 + l16][kb + 1];
            acc[t] = __builtin_amdgcn_wmma_f32_16x16x4_f32(
                /*neg_a=*/false, a, /*neg_b=*/false, b,
                /*c_mod=*/(short)0, acc[t], /*reuse_a=*/false, /*reuse_b=*/false);
        }
    }

    const int col = col0 + l16;
#pragma unroll
    for (int t = 0; t < RT; ++t) {
#pragma unroll
        for (int i = 0; i < 8; ++i) {
            int r = row0 + t * 16 + i + 8 * half;
            float s = rsqrtf(fmaxf(outdeg[r], 1.0f));
            H[r * FD + col] = acc[t][i] * s;
        }
    }
}

// -------------------------------------------------------------- edge scatter
// One wave per edge: lane c handles floats [4c, 4c+4) of the 128-wide row.
__global__ void scatter_kernel(const float* __restrict__ H, const int* __restrict__ src,
                               const int* __restrict__ dst, float* __restrict__ agg, int ne) {
    long long t = (long long)blockIdx.x * blockDim.x + threadIdx.x;
    int e = (int)(t >> 5);
    if (e >= ne) return;
    int c = ((int)t & 31) * 4;
    int s = src[e], d = dst[e];
    const float4 v = *(const float4*)(H + (long long)s * FD + c);
    float* o = agg + (long long)d * FD + c;
    atomicAdd(o + 0, v.x);
    atomicAdd(o + 1, v.y);
    atomicAdd(o + 2, v.z);
    atomicAdd(o + 3, v.w);
}

// acc += agg * rsqrt(max(indeg,1)) + bias
__global__ void accum_kernel(const float* __restrict__ agg, const float* __restrict__ indeg,
                             const float* __restrict__ bias, float* __restrict__ acc, int n_dst) {
    int i = blockIdx.x * blockDim.x + threadIdx.x;
    if (i < n_dst * FD) {
        int r = i >> 7, f = i & 127;
        acc[i] += agg[i] * rsqrtf(fmaxf(indeg[r], 1.0f)) + bias[f];
    }
}

// per-destination-type mean + ReLU (node: /1, inst: /3, svc: /2)
__global__ void finalize_kernel(float* __restrict__ out, int n_total) {
    int i = blockIdx.x * blockDim.x + threadIdx.x;
    if (i < n_total) {
        int row = i >> 7;
        float s = (row < 10000) ? 1.0f : ((row < 60000) ? (1.0f / 3.0f) : 0.5f);
        float v = out[i] * s;
        out[i] = v > 0.0f ? v : 0.0f;
    }
}

// ---------------------------------------------------------------- launcher
extern "C" void kernel_launch(void* const* d_in, const int* in_sizes, int n_in,
                              void* d_out, int out_size, void* d_ws, size_t ws_size,
                              hipStream_t stream) {
    (void)in_sizes; (void)n_in; (void)out_size; (void)ws_size;

    const float* node = (const float*)d_in[0];   // [10000,128]
    const float* inst = (const float*)d_in[1];   // [50000,128]
    const float* svc  = (const float*)d_in[2];   // [20000,128]

    struct Rel { const float* x; int n_src; int n_dst; int out_off; int base; };
    // setup_inputs order: sc, in, ni, ii, si, is ; out rows: node[0,10000) inst[10000,60000) svc[60000,80000)
    const Rel rels[6] = {
        { svc,  20000, 20000, 60000,  3 },  // sc : svc -> svc
        { inst, 50000, 10000,     0,  7 },  // in : inst -> node
        { node, 10000, 50000, 10000, 11 },  // ni : node -> inst
        { inst, 50000, 50000, 10000, 15 },  // ii : inst -> inst
        { svc,  20000, 50000, 10000, 19 },  // si : svc -> inst
        { inst, 50000, 20000, 60000, 23 },  // is : inst -> svc
    };

    const int E = 200000;
    float* ws     = (float*)d_ws;
    float* H      = ws;                         // 50000*128
    float* agg    = H   + 50000 * FD;           // 50000*128
    float* outdeg = agg + 50000 * FD;           // 50000
    float* indeg  = outdeg + 50000;             // 50000
    float* out    = (float*)d_out;              // 80000*128, used as accumulator

    const int n_total = 80000 * FD;
    zero_kernel<<<(n_total + 255) / 256, 256, 0, stream>>>(out, n_total);

    for (int r = 0; r < 6; ++r) {
        const Rel& R = rels[r];
        const int*   esrc = (const int*)d_in[R.base + 0];
        const int*   edst = (const int*)d_in[R.base + 1];
        const float* Wr   = (const float*)d_in[R.base + 2];
        const float* br   = (const float*)d_in[R.base + 3];

        zero_kernel<<<(R.n_src + 255) / 256, 256, 0, stream>>>(outdeg, R.n_src);
        zero_kernel<<<(R.n_dst + 255) / 256, 256, 0, stream>>>(indeg, R.n_dst);
        degree_kernel<<<(E + 255) / 256, 256, 0, stream>>>(esrc, edst, outdeg, indeg, E);

        gemm_outdeg_kernel<<<R.n_src / RB, 256, 0, stream>>>(R.x, Wr, outdeg, H);

        int na = R.n_dst * FD;
        zero_kernel<<<(na + 255) / 256, 256, 0, stream>>>(agg, na);

        long long st = (long long)E * 32;
        scatter_kernel<<<(int)((st + 255) / 256), 256, 0, stream>>>(H, esrc, edst, agg, E);

        accum_kernel<<<(na + 255) / 256, 256, 0, stream>>>(
            agg, indeg, br, out + (long long)R.out_off * FD, R.n_dst);
    }

    finalize_kernel<<<(n_total + 255) / 256, 256, 0, stream>>>(out, n_total);
}